// BB_DepthwiseAndSpatialSeparableConvolution_76020921139419
// MI455X (gfx1250) — compile-verified
//
#include <hip/hip_runtime.h>
#include <hip/hip_bf16.h>

typedef __attribute__((ext_vector_type(2))) float f32x2;
typedef __attribute__((ext_vector_type(8))) float f32x8;

#define BN_   4
#define C0_   36      // expand input channels
#define CH_   72      // working channels
#define H_    256
#define W_    256
#define HW_   65536
#define NPC_  262144               // B*H*W elements per channel (BN population)
#define NTOT_ 18874368             // B*CH*H*W
#define MT_   65536                // spatial M-tiles of 16  (B*H*W/4, i.e. 4x B*H*W/16)
#define EPS_  1e-3f
#define NTILES_ 5                  // ceil(72/16)
#define NPAD_   80                 // 5*16

__device__ __forceinline__ float clip8(float v) { return fminf(fmaxf(v, -8.f), 8.f); }

__device__ __forceinline__ float wave_red_sum(float v)
{
    v += __shfl_xor(v, 16, 32);
    v += __shfl_xor(v, 8, 32);
    v += __shfl_xor(v, 4, 32);
    v += __shfl_xor(v, 2, 32);
    v += __shfl_xor(v, 1, 32);
    return v;
}

// ---------------------------------------------------------------------------
// Pointwise channel-mix GEMM via V_WMMA_F32_16X16X4_F32.
// raw = W * act(in) + bias ; writes raw, accumulates per-channel sum/sumsq.
// One wave = one 16(M=spatial) x 72(N, 5 tiles) strip: the A-frag is loaded
// from global ONCE and fed to 5 independent WMMAs whose B-frags come from a
// zero-padded LDS copy of the weights (conflict-free padded stride).
// ACT: 0 = identity input (expand), 2 = BN + clip input.
// ---------------------------------------------------------------------------
template <int ACT, int CIN>
__global__ __launch_bounds__(128) void pw_wmma_kernel(
    const float* __restrict__ in,
    const float* __restrict__ wmat,   // [CH_ x CIN] row-major (o,c)
    const float* __restrict__ bias,   // [CH_]
    const float* __restrict__ sc, const float* __restrict__ sh,
    float* __restrict__ out,          // [B, CH_, H, W] raw
    float* __restrict__ ssum, float* __restrict__ ssq)
{
    constexpr int LWS = (CIN == 72) ? 76 : 42;   // padded LDS row stride (even, bank-safe)
    __shared__ float lw[NPAD_ * LWS];
    __shared__ float lsc[CH_];
    __shared__ float lsh[CH_];
    __shared__ float s_sum[CH_];
    __shared__ float s_sq[CH_];

    const int tid = threadIdx.x;

    // cooperative stage: weights (zero-padded rows 72..79), input-BN params, stats
    for (int i = tid; i < NPAD_ * CIN; i += 128) {
        const int n = i / CIN;
        const int k = i - n * CIN;
        lw[n * LWS + k] = (n < CH_) ? wmat[n * CIN + k] : 0.f;
    }
    if (tid < CH_) {
        s_sum[tid] = 0.f;
        s_sq[tid]  = 0.f;
        if (ACT == 2) { lsc[tid] = sc[tid]; lsh[tid] = sh[tid]; }
    }
    __syncthreads();

    const int wave    = tid >> 5;
    const int lane    = tid & 31;
    const int laneRow = lane & 15;
    const int khalf   = (lane >> 4) << 1;          // 0 or 2
    const int m0      = (blockIdx.x * 4 + wave) * 16;
    const int b_idx   = m0 >> 16;                  // HW_ == 65536
    const int hwbase  = m0 & 0xFFFF;

    const float* inb = in + (size_t)b_idx * CIN * HW_ + hwbase + laneRow;

    f32x8 acc[NTILES_];
#pragma unroll
    for (int nt = 0; nt < NTILES_; ++nt)
        acc[nt] = f32x8{0.f, 0.f, 0.f, 0.f, 0.f, 0.f, 0.f, 0.f};

    for (int k0 = 0; k0 < CIN; k0 += 4) {
        const int kA = k0 + khalf;
        float x0 = inb[(size_t)kA * HW_];
        float x1 = inb[(size_t)(kA + 1) * HW_];
        if (ACT == 2) {
            x0 = clip8(x0 * lsc[kA]     + lsh[kA]);
            x1 = clip8(x1 * lsc[kA + 1] + lsh[kA + 1]);
        }
        const f32x2 afrag = {x0, x1};
#pragma unroll
        for (int nt = 0; nt < NTILES_; ++nt) {
            const float* wp = &lw[(nt * 16 + laneRow) * LWS + kA];
            const f32x2 bfrag = {wp[0], wp[1]};
            acc[nt] = __builtin_amdgcn_wmma_f32_16x16x4_f32(
                false, afrag, false, bfrag, (short)0, acc[nt], false, false);
        }
    }

#pragma unroll
    for (int nt = 0; nt < NTILES_; ++nt) {
        const int n = nt * 16 + laneRow;
        if (n < CH_) {
            const float bb = bias[n];
            float vr[8];
            float lsum = 0.f, lsq = 0.f;
#pragma unroll
            for (int r = 0; r < 8; ++r) {
                const float v = acc[nt][r] + bb;   // raw pre-BN, row M = r + 8*(lane>=16)
                vr[r] = v;
                lsum += v; lsq += v * v;
            }
            float* outb = out + ((size_t)b_idx * CH_ + n) * HW_ + hwbase + ((lane >> 4) << 3);
            float4 t0; t0.x = vr[0]; t0.y = vr[1]; t0.z = vr[2]; t0.w = vr[3];
            float4 t1; t1.x = vr[4]; t1.y = vr[5]; t1.z = vr[6]; t1.w = vr[7];
            *(float4*)(outb)     = t0;
            *(float4*)(outb + 4) = t1;
            atomicAdd(&s_sum[n], lsum);
            atomicAdd(&s_sq[n],  lsq);
        }
    }
    __syncthreads();
    if (tid < CH_) {
        atomicAdd(&ssum[tid], s_sum[tid]);
        atomicAdd(&ssq[tid],  s_sq[tid]);
    }
}

// ---------------------------------------------------------------------------
// IIR along H: y0=(a+b+c)x0 ; y[h]=a*x[h]+b*x[h-1]+c*y[h-1].
// One thread per (b,c,w) column, block = one (b,c) plane, W threads.
// ACT: 0 = identity input, 1 = BN+ReLU input.
// ---------------------------------------------------------------------------
template <int ACT>
__global__ __launch_bounds__(256) void iir_kernel(
    const float* __restrict__ in,
    const float* __restrict__ coef,   // [3][CH_]  (a,b,c)
    const float* __restrict__ sc, const float* __restrict__ sh,
    float* __restrict__ out,
    float* __restrict__ ssum, float* __restrict__ ssq)
{
    __shared__ float s_sum, s_sq;
    const int w  = threadIdx.x;
    const int bc = blockIdx.x;            // b*CH_ + c
    const int c  = bc % CH_;
    if (w == 0) { s_sum = 0.f; s_sq = 0.f; }
    __syncthreads();

    const float ka = coef[c], kb = coef[CH_ + c], kc = coef[2 * CH_ + c];
    float s = 1.f, t = 0.f;
    if (ACT == 1) { s = sc[c]; t = sh[c]; }

    const float* pin  = in  + (size_t)bc * HW_ + w;
    float*       pout = out + (size_t)bc * HW_ + w;

    float xprev = pin[0];
    if (ACT == 1) xprev = fmaxf(xprev * s + t, 0.f);
    float y = (ka + kb + kc) * xprev;
    pout[0] = y;
    float lsum = y, lsq = y * y;

    for (int h = 1; h < H_; ++h) {
        float x = pin[(size_t)h * W_];
        if (ACT == 1) x = fmaxf(x * s + t, 0.f);
        y = ka * x + kb * xprev + kc * y;
        xprev = x;
        pout[(size_t)h * W_] = y;
        lsum += y; lsq += y * y;
    }
    lsum = wave_red_sum(lsum);
    lsq  = wave_red_sum(lsq);
    if ((w & 31) == 0) { atomicAdd(&s_sum, lsum); atomicAdd(&s_sq, lsq); }
    __syncthreads();
    if (w == 0) { atomicAdd(&ssum[c], s_sum); atomicAdd(&ssq[c], s_sq); }
}

// ---------------------------------------------------------------------------
// Depthwise 1x5 along W (pad 2); input gets BN+clip of previous IIR stage.
// Block = one (b,c,h) row of 256, staged through LDS with halo.
// ---------------------------------------------------------------------------
__global__ __launch_bounds__(256) void dw_kernel(
    const float* __restrict__ in,
    const float* __restrict__ wk,     // [CH_][5]
    const float* __restrict__ bias,   // [CH_]
    const float* __restrict__ sc, const float* __restrict__ sh,
    float* __restrict__ out,
    float* __restrict__ ssum, float* __restrict__ ssq)
{
    __shared__ float row[W_ + 4];
    __shared__ float s_sum, s_sq;
    const int w   = threadIdx.x;
    const int idx = blockIdx.x;           // ((b*CH_ + c)*H_ + h)
    const int c   = (idx / H_) % CH_;
    if (w == 0) {
        s_sum = 0.f; s_sq = 0.f;
        row[0] = row[1] = 0.f;
        row[W_ + 2] = row[W_ + 3] = 0.f;
    }
    const size_t base = (size_t)idx * W_;
    float x = in[base + w];
    x = clip8(x * sc[c] + sh[c]);
    row[w + 2] = x;
    __syncthreads();

    const float* k = wk + c * 5;
    const float y = k[0] * row[w] + k[1] * row[w + 1] + k[2] * row[w + 2] +
                    k[3] * row[w + 3] + k[4] * row[w + 4] + bias[c];
    out[base + w] = y;
    float lsum = wave_red_sum(y);
    float lsq  = wave_red_sum(y * y);
    if ((w & 31) == 0) { atomicAdd(&s_sum, lsum); atomicAdd(&s_sq, lsq); }
    __syncthreads();
    if (w == 0) { atomicAdd(&ssum[c], s_sum); atomicAdd(&ssq[c], s_sq); }
}

// ---------------------------------------------------------------------------
// dst = base(optionally BN+ReLU) + ReLU(add * s + t)
// ---------------------------------------------------------------------------
template <int BASE_ACT>
__global__ __launch_bounds__(256) void residual_kernel(
    const float* __restrict__ base,
    const float* __restrict__ add,
    const float* __restrict__ sc_b, const float* __restrict__ sh_b,
    const float* __restrict__ sc_a, const float* __restrict__ sh_a,
    float* __restrict__ dst)
{
    const size_t i = (size_t)blockIdx.x * 256 + threadIdx.x;
    const int c = (int)((i >> 16) % CH_);
    float bv = base[i];
    if (BASE_ACT) bv = fmaxf(bv * sc_b[c] + sh_b[c], 0.f);
    const float av = fmaxf(add[i] * sc_a[c] + sh_a[c], 0.f);
    dst[i] = bv + av;
}

// Per-channel BN stats -> affine scale/shift:  y_norm = y*s + t
__global__ void finalize_kernel(const float* __restrict__ ssum,
                                const float* __restrict__ ssq,
                                const float* __restrict__ gamma,
                                const float* __restrict__ beta,
                                float* __restrict__ st_s,
                                float* __restrict__ st_t)
{
    const int c = threadIdx.x;
    if (c >= CH_) return;
    const float mean = ssum[c] * (1.f / NPC_);
    const float var  = ssq[c] * (1.f / NPC_) - mean * mean;
    const float s    = gamma[c] * rsqrtf(var + EPS_);
    st_s[c] = s;
    st_t[c] = beta[c] - mean * s;
}

__global__ void zero_kernel(float* __restrict__ p, int n)
{
    const int i = blockIdx.x * 256 + threadIdx.x;
    if (i < n) p[i] = 0.f;
}

// ---------------------------------------------------------------------------
extern "C" void kernel_launch(void* const* d_in, const int* in_sizes, int n_in,
                              void* d_out, int out_size, void* d_ws, size_t ws_size,
                              hipStream_t stream)
{
    const float* x      = (const float*)d_in[0];
    const float* ex_w   = (const float*)d_in[1];
    const float* ex_b   = (const float*)d_in[2];
    const float* ex_g   = (const float*)d_in[3];
    const float* ex_be  = (const float*)d_in[4];
    const float* iir_w  = (const float*)d_in[5];   // (4,3,72)
    const float* iir_g  = (const float*)d_in[6];   // (4,72)
    const float* iir_be = (const float*)d_in[7];
    const float* dw_w   = (const float*)d_in[8];   // (4,72,5)
    const float* dw_b   = (const float*)d_in[9];
    const float* dw_g   = (const float*)d_in[10];
    const float* dw_be  = (const float*)d_in[11];
    const float* pw_w   = (const float*)d_in[12];  // (4,72,72)
    const float* pw_b   = (const float*)d_in[13];
    const float* pw_g   = (const float*)d_in[14];
    const float* pw_be  = (const float*)d_in[15];
    float* outp = (float*)d_out;

    // Workspace layout (floats)
    float* ws   = (float*)d_ws;
    float* ssum = ws;                 // 13*72
    float* ssq  = ssum + 13 * CH_;    // 13*72
    float* st_s = ssq  + 13 * CH_;    // 13*72
    float* st_t = st_s + 13 * CH_;    // 13*72
    float* y1   = st_t + 13 * CH_;    // NTOT_  (also holds residual h1)
    float* bufA = y1 + (size_t)NTOT_;
    float* bufB = bufA + (size_t)NTOT_;

#define SUM(s) (ssum + (s) * CH_)
#define SQ(s)  (ssq  + (s) * CH_)
#define SS(s)  (st_s + (s) * CH_)
#define TT(s)  (st_t + (s) * CH_)

    zero_kernel<<<(13 * CH_ * 2 + 255) / 256, 256, 0, stream>>>(ssum, 13 * CH_ * 2);

    const dim3 gpw(MT_ / 4);          // 4 waves/block, each wave: 16 M x 72 N strip
    const int  giir = BN_ * CH_;
    const int  gdw  = BN_ * CH_ * H_;
    const int  gres = NTOT_ / 256;

    // ---- stage 0: expand 36 -> 72 (WMMA), raw + stats ----
    pw_wmma_kernel<0, C0_><<<gpw, 128, 0, stream>>>(x, ex_w, ex_b, nullptr, nullptr,
                                                    y1, SUM(0), SQ(0));
    finalize_kernel<<<1, 128, 0, stream>>>(SUM(0), SQ(0), ex_g, ex_be, SS(0), TT(0));

    // ---- unit 0 : in = BN+ReLU(y1) ----
    iir_kernel<1><<<giir, 256, 0, stream>>>(y1, iir_w + 0 * 216, SS(0), TT(0),
                                            bufA, SUM(1), SQ(1));
    finalize_kernel<<<1, 128, 0, stream>>>(SUM(1), SQ(1), iir_g + 0, iir_be + 0, SS(1), TT(1));
    dw_kernel<<<gdw, 256, 0, stream>>>(bufA, dw_w + 0, dw_b + 0, SS(1), TT(1),
                                       bufB, SUM(2), SQ(2));
    finalize_kernel<<<1, 128, 0, stream>>>(SUM(2), SQ(2), dw_g + 0, dw_be + 0, SS(2), TT(2));
    pw_wmma_kernel<2, CH_><<<gpw, 128, 0, stream>>>(bufB, pw_w + 0, pw_b + 0, SS(2), TT(2),
                                                    bufA, SUM(3), SQ(3));
    finalize_kernel<<<1, 128, 0, stream>>>(SUM(3), SQ(3), pw_g + 0, pw_be + 0, SS(3), TT(3));

    // ---- unit 1 : in = BN+ReLU(p0 raw) ----
    iir_kernel<1><<<giir, 256, 0, stream>>>(bufA, iir_w + 1 * 216, SS(3), TT(3),
                                            bufB, SUM(4), SQ(4));
    finalize_kernel<<<1, 128, 0, stream>>>(SUM(4), SQ(4), iir_g + 72, iir_be + 72, SS(4), TT(4));
    dw_kernel<<<gdw, 256, 0, stream>>>(bufB, dw_w + 360, dw_b + 72, SS(4), TT(4),
                                       bufA, SUM(5), SQ(5));
    finalize_kernel<<<1, 128, 0, stream>>>(SUM(5), SQ(5), dw_g + 72, dw_be + 72, SS(5), TT(5));
    pw_wmma_kernel<2, CH_><<<gpw, 128, 0, stream>>>(bufA, pw_w + 5184, pw_b + 72, SS(5), TT(5),
                                                    bufB, SUM(6), SQ(6));
    finalize_kernel<<<1, 128, 0, stream>>>(SUM(6), SQ(6), pw_g + 72, pw_be + 72, SS(6), TT(6));

    // ---- residual 1: h1 = ReLU(BN0(y1)) + ReLU(BN6(p1))  (in place into y1) ----
    residual_kernel<1><<<gres, 256, 0, stream>>>(y1, bufB, SS(0), TT(0), SS(6), TT(6), y1);

    // ---- unit 2 : in = h1 (already activated) ----
    iir_kernel<0><<<giir, 256, 0, stream>>>(y1, iir_w + 2 * 216, nullptr, nullptr,
                                            bufA, SUM(7), SQ(7));
    finalize_kernel<<<1, 128, 0, stream>>>(SUM(7), SQ(7), iir_g + 144, iir_be + 144, SS(7), TT(7));
    dw_kernel<<<gdw, 256, 0, stream>>>(bufA, dw_w + 720, dw_b + 144, SS(7), TT(7),
                                       bufB, SUM(8), SQ(8));
    finalize_kernel<<<1, 128, 0, stream>>>(SUM(8), SQ(8), dw_g + 144, dw_be + 144, SS(8), TT(8));
    pw_wmma_kernel<2, CH_><<<gpw, 128, 0, stream>>>(bufB, pw_w + 2 * 5184, pw_b + 144, SS(8), TT(8),
                                                    bufA, SUM(9), SQ(9));
    finalize_kernel<<<1, 128, 0, stream>>>(SUM(9), SQ(9), pw_g + 144, pw_be + 144, SS(9), TT(9));

    // ---- unit 3 ----
    iir_kernel<1><<<giir, 256, 0, stream>>>(bufA, iir_w + 3 * 216, SS(9), TT(9),
                                            bufB, SUM(10), SQ(10));
    finalize_kernel<<<1, 128, 0, stream>>>(SUM(10), SQ(10), iir_g + 216, iir_be + 216, SS(10), TT(10));
    dw_kernel<<<gdw, 256, 0, stream>>>(bufB, dw_w + 1080, dw_b + 216, SS(10), TT(10),
                                       bufA, SUM(11), SQ(11));
    finalize_kernel<<<1, 128, 0, stream>>>(SUM(11), SQ(11), dw_g + 216, dw_be + 216, SS(11), TT(11));
    pw_wmma_kernel<2, CH_><<<gpw, 128, 0, stream>>>(bufA, pw_w + 3 * 5184, pw_b + 216, SS(11), TT(11),
                                                    bufB, SUM(12), SQ(12));
    finalize_kernel<<<1, 128, 0, stream>>>(SUM(12), SQ(12), pw_g + 216, pw_be + 216, SS(12), TT(12));

    // ---- final: out = h1 + ReLU(BN12(p3)) ----
    residual_kernel<0><<<gres, 256, 0, stream>>>(y1, bufB, nullptr, nullptr, SS(12), TT(12), outp);

#undef SUM
#undef SQ
#undef SS
#undef TT
}